// FaceRenderer_23381801959549
// MI455X (gfx1250) — compile-verified
//
#include <hip/hip_runtime.h>
#include <stdint.h>

#define IM       224
#define HW       (IM * IM)
#define TCHUNK   256          // triangles per LDS chunk
#define NSLICE   8            // triangle slices (grid.y) for occupancy
#define EPSF     1e-8f
#define NEARF    0.01f
// Baked projection*view matrix (eye=(0,0,10), fov from 112/1015):
// clip.x = FC*x ; clip.y = FC*y ; clip.z = C22*z + M23 ; clip.w = 10 - z
#define FCONST   9.0625f
#define C22v     (-1.0004000800160032f)
#define M23v     (9.983996799359872f)

typedef unsigned int       u32;
typedef unsigned long long u64;
typedef u32   u32x4v __attribute__((ext_vector_type(4)));
typedef int   i32x4v __attribute__((ext_vector_type(4)));
typedef int   i32x8v __attribute__((ext_vector_type(8)));
typedef float v2f    __attribute__((ext_vector_type(2)));
typedef float v8f    __attribute__((ext_vector_type(8)));

// ---------------- Tensor Data Mover staging (CDNA5) ----------------
// Copies nelem8 * 8 bytes from global gaddr into LDS at lds_off using the TDM.
// D# packing per CDNA5 ISA.
__device__ __forceinline__ void tdm_issue(u32 lds_off, u64 gaddr, u32 nelem8) {
#if defined(__HIP_DEVICE_COMPILE__) && __has_builtin(__builtin_amdgcn_tensor_load_to_lds)
  u32x4v g0;
  g0.x = 1u;                                                  // count=1, user-mode
  g0.y = lds_off;                                             // LDS byte address
  g0.z = (u32)gaddr;                                          // global_addr[31:0]
  g0.w = ((u32)(gaddr >> 32) & 0x01FFFFFFu) | (2u << 30);     // addr[56:32] | type=2
  i32x8v g1;
  g1[0] = (int)(3u << 16);                                    // data_size = 8B
  g1[1] = (int)((nelem8 & 0xFFFFu) << 16);                    // tensor_dim0[15:0]
  g1[2] = (int)(((nelem8 >> 16) & 0xFFFFu) | (1u << 16));     // tdim0[31:16] | tensor_dim1=1
  g1[3] = (int)((nelem8 & 0xFFFFu) << 16);                    // tile_dim0
  g1[4] = 1;                                                  // tile_dim1=1, tile_dim2=0
  g1[5] = (int)nelem8;                                        // tensor_dim0_stride[31:0]
  g1[6] = 0;
  g1[7] = 0;
  i32x4v z4 = {0, 0, 0, 0};
#if __clang_major__ >= 23
  i32x8v z8 = {0, 0, 0, 0, 0, 0, 0, 0};
  __builtin_amdgcn_tensor_load_to_lds(g0, g1, z4, z4, z8, 0);
#else
  __builtin_amdgcn_tensor_load_to_lds(g0, g1, z4, z4, 0);
#endif
#endif
}

// Stage one 256-triangle chunk (256*48 = 12288 B) into LDS. Each of the 8
// waves issues a TDM descriptor for its own 1536 B (192 x 8B) slice.
__device__ __forceinline__ void stage_chunk(float* ldsDst, const float* gsrc, int tid) {
#if defined(__HIP_DEVICE_COMPILE__) && __has_builtin(__builtin_amdgcn_tensor_load_to_lds)
  u32 wave    = (u32)(__builtin_amdgcn_readfirstlane(tid) >> 5);
  u32 lds_off = (u32)(uintptr_t)ldsDst + wave * 1536u;
  u64 ga      = (u64)(uintptr_t)gsrc + (u64)wave * 1536u;
  tdm_issue(lds_off, ga, 192u);
#else
  const float4* s = (const float4*)gsrc;
  float4*       d = (float4*)ldsDst;
  for (int i = tid; i < (TCHUNK * 12) / 4; i += 256) d[i] = s[i];
#endif
}

__device__ __forceinline__ void stage_wait() {
#if defined(__HIP_DEVICE_COMPILE__) && __has_builtin(__builtin_amdgcn_tensor_load_to_lds)
  __builtin_amdgcn_s_wait_tensorcnt(0);
#endif
  __syncthreads();
}

__device__ __forceinline__ u64 shfl_xor_u64_w16(u64 v, int mask) {
  int lo = __shfl_xor((int)(u32)v, mask, 16);
  int hi = __shfl_xor((int)(u32)(v >> 32), mask, 16);
  return ((u64)(u32)hi << 32) | (u32)lo;
}

// ---------------- Kernels ----------------
__global__ void init_keys(u64* __restrict__ keys) {
  int p = blockIdx.x * 256 + threadIdx.x;
  if (p < HW) keys[p] = ~0ull;
}

// Per-triangle setup: area-normalized edge coefficients (so inside-test is
// b_e >= 0, matching ref's w_e * sign(area) >= 0) + affine z row, and an
// epilogue record (inv_w + vertex indices).
__global__ __launch_bounds__(256) void setup_kernel(
    const float* __restrict__ verts, const int* __restrict__ tris,
    float* __restrict__ inner, float* __restrict__ epi, int nt, int ntp, int nv)
{
  int t = blockIdx.x * 256 + threadIdx.x;
  if (t >= ntp) return;
  bool valid = (t < nt);
  int idx[3] = {0, 0, 0};
  if (valid) {
    idx[0] = tris[t * 3 + 0]; idx[1] = tris[t * 3 + 1]; idx[2] = tris[t * 3 + 2];
#pragma unroll
    for (int k = 0; k < 3; ++k) idx[k] = min(max(idx[k], 0), nv - 1);
  }
  float X[3], Y[3], Z[3], IW[3], WV[3];
#pragma unroll
  for (int k = 0; k < 3; ++k) {
    float x = verts[idx[k] * 3 + 0];
    float y = verts[idx[k] * 3 + 1];
    float z = verts[idx[k] * 3 + 2];
    float w  = 10.0f - z;                       // clip.w
    float ws = (fabsf(w) > EPSF) ? w : EPSF;
    float iw = 1.0f / ws;
    X[k] = FCONST * x * iw;
    Y[k] = FCONST * y * iw;
    Z[k] = fmaf(C22v, z, M23v) * iw;
    IW[k] = iw; WV[k] = w;
  }
  // edge(a,b)(p) = d.x*(p.y-a.y) - d.y*(p.x-a.x) = A*px + B*py + C
  float d0x = X[2] - X[1], d0y = Y[2] - Y[1];   // w0: a=v1,b=v2
  float d1x = X[0] - X[2], d1y = Y[0] - Y[2];   // w1: a=v2,b=v0
  float d2x = X[1] - X[0], d2y = Y[1] - Y[0];   // w2: a=v0,b=v1
  float A0 = -d0y, B0 = d0x, Cc0 = d0y * X[1] - d0x * Y[1];
  float A1 = -d1y, B1 = d1x, Cc1 = d1y * X[2] - d1x * Y[2];
  float A2 = -d2y, B2 = d2x, Cc2 = d2y * X[0] - d2x * Y[0];
  float area = d2x * (Y[2] - Y[0]) - d2y * (X[2] - X[0]);  // e1 x e2
  valid = valid && (fabsf(area) > EPSF) &&
          (WV[0] > NEARF) && (WV[1] > NEARF) && (WV[2] > NEARF);
  float r[12];
  if (valid) {
    float ia = 1.0f / area;
    r[0] = A0 * ia; r[1] = B0 * ia; r[2] = Cc0 * ia;
    r[3] = A1 * ia; r[4] = B1 * ia; r[5] = Cc1 * ia;
    r[6] = A2 * ia; r[7] = B2 * ia; r[8] = Cc2 * ia;
    r[9]  = r[0] * Z[0] + r[3] * Z[1] + r[6] * Z[2];   // Za
    r[10] = r[1] * Z[0] + r[4] * Z[1] + r[7] * Z[2];   // Zb
    r[11] = r[2] * Z[0] + r[5] * Z[1] + r[8] * Z[2];   // Zc
  } else {
    r[0] = r[1] = r[3] = r[4] = r[6] = r[7] = 0.0f;
    r[2] = r[5] = r[8] = -1.0f;                        // never inside
    r[9] = r[10] = r[11] = 0.0f;
  }
  float4* ip = (float4*)(inner + (size_t)t * 12);
  ip[0] = make_float4(r[0], r[1], r[2],  r[3]);
  ip[1] = make_float4(r[4], r[5], r[6],  r[7]);
  ip[2] = make_float4(r[8], r[9], r[10], r[11]);
  float* e = epi + (size_t)t * 8;
  e[0] = IW[0]; e[1] = IW[1]; e[2] = IW[2];
  e[3] = __int_as_float(idx[0]);
  e[4] = __int_as_float(idx[1]);
  e[5] = __int_as_float(idx[2]);
  e[6] = 0.0f; e[7] = 0.0f;
}

// Rasterize: grid = (196 pixel tiles, NSLICE triangle slices). TDM double-
// buffers triangle records through LDS. The 4 per-pixel affine quantities
// (3 edges + z) are evaluated with V_WMMA_F32_16X16X4_F32:
//   A (16x4) = [px[m], py, 1, 0] per pixel row  (lanes 0-15: K0/K1, 16-31: K2/K3)
//   B (4x16) = [nA, nB, nC, 0]  per triangle col (lanes 0-15: K0/K1, 16-31: K2/K3)
//   D (16x16): reg r <-> pixel m = r + 8*(lane>=16), col n = lane%16.
// One WMMA per quantity keeps the 4 D results element-aligned, so the
// inside/z tests stay elementwise. Per-lane running best is merged at the end
// with a 16-lane key min-reduction + global atomicMin on (sortable_z | idx),
// exactly reproducing "min z, tie -> lowest triangle index".
__global__ __launch_bounds__(256) void raster_kernel(
    const float* __restrict__ inner, u64* __restrict__ keys, int nch)
{
  __shared__ __align__(16) float lds[2][TCHUNK * 12];   // 24 KB double buffer
  const int tid = threadIdx.x;
  const int bx  = blockIdx.x;            // 0..195 pixel tiles (14x14 of 16x16)
  const int s   = blockIdx.y;            // triangle slice
  const int tx  = bx % (IM / 16), ty = bx / (IM / 16);

#if defined(__HIP_DEVICE_COMPILE__) && __has_builtin(__builtin_amdgcn_wmma_f32_16x16x4_f32)
  const int lane = tid & 31;
  const int wav  = tid >> 5;
  const int n    = lane & 15;            // triangle column / pixel row id
  const int half = lane >> 4;
  // Wave covers 2 image rows (M-tiles): iy0 = ty*16 + wav*2 (+mt)
  const int   iy0 = ty * 16 + wav * 2;
  const float pxm = (tx * 16 + n + 0.5f) * (2.0f / IM) - 1.0f;
  v2f A[2];
#pragma unroll
  for (int mt = 0; mt < 2; ++mt) {
    float py = 1.0f - (iy0 + mt + 0.5f) * (2.0f / IM);
    A[mt].x = half ? 1.0f : pxm;     // K0 | K2
    A[mt].y = half ? 0.0f : py;      // K1 | K3
  }
  float bestZ[2][8];
  int   bestT[2][8];
#pragma unroll
  for (int mt = 0; mt < 2; ++mt)
#pragma unroll
    for (int r = 0; r < 8; ++r) { bestZ[mt][r] = __builtin_inff(); bestT[mt][r] = -1; }

  if (s < nch) stage_chunk(lds[0], inner + (size_t)s * TCHUNK * 12, tid);
  int k = 0;
  for (int c = s; c < nch; c += NSLICE, ++k) {
    int cur = k & 1;
    stage_wait();
    int c2 = c + NSLICE;
    if (c2 < nch)
      stage_chunk(lds[cur ^ 1], inner + (size_t)c2 * TCHUNK * 12, tid);
    const float* buf = lds[cur];
    const int tbase = c * TCHUNK;
    for (int g = 0; g < TCHUNK / 16; ++g) {
      const int rec = (g * 16 + n) * 12;   // conflict-free: 12n mod 64 distinct
      v2f B[4];
#pragma unroll
      for (int q = 0; q < 4; ++q) {        // q: b0,b1,b2,z coefficient rows
        int base = rec + q * 3;
        float lo01 = buf[base + (half ? 2 : 0)];
        float hi01 = buf[base + 1];
        B[q].x = lo01;                     // K0 | K2 (nA | nC)
        B[q].y = half ? 0.0f : hi01;       // K1 | K3 (nB | 0)
      }
      const int tg = tbase + g * 16 + n;
      v8f cz = {0, 0, 0, 0, 0, 0, 0, 0};
#pragma unroll
      for (int mt = 0; mt < 2; ++mt) {
        v8f d0 = __builtin_amdgcn_wmma_f32_16x16x4_f32(false, A[mt], false, B[0], (short)0, cz, false, false);
        v8f d1 = __builtin_amdgcn_wmma_f32_16x16x4_f32(false, A[mt], false, B[1], (short)0, cz, false, false);
        v8f d2 = __builtin_amdgcn_wmma_f32_16x16x4_f32(false, A[mt], false, B[2], (short)0, cz, false, false);
        v8f dz = __builtin_amdgcn_wmma_f32_16x16x4_f32(false, A[mt], false, B[3], (short)0, cz, false, false);
#pragma unroll
        for (int r = 0; r < 8; ++r) {
          float zv = dz[r];
          bool ins = (d0[r] >= 0.0f) & (d1[r] >= 0.0f) & (d2[r] >= 0.0f) &
                     (zv >= -1.0f) & (zv <= 1.0f) & (zv < bestZ[mt][r]);
          if (ins) { bestZ[mt][r] = zv; bestT[mt][r] = tg; }
        }
      }
    }
  }
  // Merge: per (mt, r), min (sortable_z | idx) across the 16 triangle columns.
#pragma unroll
  for (int mt = 0; mt < 2; ++mt) {
#pragma unroll
    for (int r = 0; r < 8; ++r) {
      u64 key = ~0ull;
      if (bestT[mt][r] >= 0) {
        u32 zb = __float_as_uint(bestZ[mt][r]);
        u32 sz = (zb & 0x80000000u) ? ~zb : (zb | 0x80000000u);
        key = ((u64)sz << 32) | (u32)bestT[mt][r];
      }
#pragma unroll
      for (int msk = 1; msk < 16; msk <<= 1) {
        u64 o = shfl_xor_u64_w16(key, msk);
        key = (o < key) ? o : key;
      }
      if (n == 0 && key != ~0ull) {
        int ix = tx * 16 + r + half * 8;   // lanes 0/16 hold halves' minima
        atomicMin(&keys[(iy0 + mt) * IM + ix], key);
      }
    }
  }
#else
  // Scalar fallback (also used for host-compile parse).
  int ix = tx * 16 + (tid & 15);
  int iy = ty * 16 + (tid >> 4);
  float px = (ix + 0.5f) * (2.0f / IM) - 1.0f;
  float py = 1.0f - (iy + 0.5f) * (2.0f / IM);
  float bestZ = __builtin_inff();
  int   bestT = -1;
  if (s < nch) stage_chunk(lds[0], inner + (size_t)s * TCHUNK * 12, tid);
  int k = 0;
  for (int c = s; c < nch; c += NSLICE, ++k) {
    int cur = k & 1;
    stage_wait();
    int c2 = c + NSLICE;
    if (c2 < nch)
      stage_chunk(lds[cur ^ 1], inner + (size_t)c2 * TCHUNK * 12, tid);
    const float* buf = lds[cur];
    int tbase = c * TCHUNK;
#pragma unroll 2
    for (int t = 0; t < TCHUNK; ++t) {
      const float4* rp = (const float4*)(buf + t * 12);
      float4 r0 = rp[0], r1 = rp[1], r2 = rp[2];
      float b0 = fmaf(r0.y, py, fmaf(r0.x, px, r0.z));
      float b1 = fmaf(r1.x, py, fmaf(r0.w, px, r1.y));
      float b2 = fmaf(r1.w, py, fmaf(r1.z, px, r2.x));
      float z  = fmaf(r2.z, py, fmaf(r2.y, px, r2.w));
      bool inside = (b0 >= 0.0f) & (b1 >= 0.0f) & (b2 >= 0.0f) &
                    (z >= -1.0f) & (z <= 1.0f) & (z < bestZ);
      if (inside) { bestZ = z; bestT = tbase + t; }
    }
  }
  if (bestT >= 0) {
    u32 zb = __float_as_uint(bestZ);
    u32 sz = (zb & 0x80000000u) ? ~zb : (zb | 0x80000000u);
    u64 key = ((u64)sz << 32) | (u32)bestT;
    atomicMin(&keys[iy * IM + ix], key);
  }
#endif
}

// Shade winners: perspective-corrected barycentrics over vertex colors.
__global__ __launch_bounds__(256) void resolve_kernel(
    const float* __restrict__ inner, const float* __restrict__ epi,
    const u64* __restrict__ keys, const float* __restrict__ colr,
    float* __restrict__ out)
{
  int p = blockIdx.x * 256 + threadIdx.x;
  if (p >= HW) return;
  u64 key = keys[p];
  float c0 = 0.0f, c1 = 0.0f, c2 = 0.0f;
  if (key != ~0ull) {
    int t  = (int)(u32)key;
    int ix = p % IM, iy = p / IM;
    float px = (ix + 0.5f) * (2.0f / IM) - 1.0f;
    float py = 1.0f - (iy + 0.5f) * (2.0f / IM);
    const float4* rp = (const float4*)(inner + (size_t)t * 12);
    float4 r0 = rp[0], r1 = rp[1], r2 = rp[2];
    float b0 = fmaf(r0.y, py, fmaf(r0.x, px, r0.z));
    float b1 = fmaf(r1.x, py, fmaf(r0.w, px, r1.y));
    float b2 = fmaf(r1.w, py, fmaf(r1.z, px, r2.x));
    const float* e = epi + (size_t)t * 8;
    float p0 = b0 * e[0], p1 = b1 * e[1], p2 = b2 * e[2];
    float ssum = p0 + p1 + p2;
    float inv  = 1.0f / ((fabsf(ssum) > EPSF) ? ssum : 1.0f);
    p0 *= inv; p1 *= inv; p2 *= inv;
    int i0 = __float_as_int(e[3]);
    int i1 = __float_as_int(e[4]);
    int i2 = __float_as_int(e[5]);
    c0 = p0 * colr[i0 * 3 + 0] + p1 * colr[i1 * 3 + 0] + p2 * colr[i2 * 3 + 0];
    c1 = p0 * colr[i0 * 3 + 1] + p1 * colr[i1 * 3 + 1] + p2 * colr[i2 * 3 + 1];
    c2 = p0 * colr[i0 * 3 + 2] + p1 * colr[i1 * 3 + 2] + p2 * colr[i2 * 3 + 2];
  }
  out[p * 3 + 0] = c0;
  out[p * 3 + 1] = c1;
  out[p * 3 + 2] = c2;
}

extern "C" void kernel_launch(void* const* d_in, const int* in_sizes, int n_in,
                              void* d_out, int out_size, void* d_ws, size_t ws_size,
                              hipStream_t stream) {
  const float* verts = (const float*)d_in[0];   // (1, NV, 3) f32
  const float* colr  = (const float*)d_in[1];   // (1, NV, 3) f32
  const int*   tris  = (const int*)d_in[2];     // (NT, 3) int
  int nv  = in_sizes[0] / 3;
  int nt  = in_sizes[2] / 3;
  int ntp = ((nt + TCHUNK - 1) / TCHUNK) * TCHUNK;
  int nch = ntp / TCHUNK;

  // Workspace layout: inner[ntp*12] f32 | epi[ntp*8] f32 | keys[HW] u64
  float* inner = (float*)d_ws;
  float* epi   = inner + (size_t)ntp * 12;
  u64*   keys  = (u64*)(epi + (size_t)ntp * 8);

  init_keys<<<(HW + 255) / 256, 256, 0, stream>>>(keys);
  setup_kernel<<<(ntp + 255) / 256, 256, 0, stream>>>(verts, tris, inner, epi, nt, ntp, nv);
  raster_kernel<<<dim3((IM / 16) * (IM / 16), NSLICE), 256, 0, stream>>>(inner, keys, nch);
  resolve_kernel<<<(HW + 255) / 256, 256, 0, stream>>>(inner, epi, keys, colr, (float*)d_out);
}